// ImageDot_30597347016804
// MI455X (gfx1250) — compile-verified
//
#include <hip/hip_runtime.h>

typedef __attribute__((ext_vector_type(4))) float f4;

#define ALPHA_C 0.9f
#define RADIUS_C 25.0f
#define HH 256
#define WW 256
#define NPIX (HH * WW)          // 65536 pixels
#define NPIX4 (NPIX / 4)        // 16384 float4 per plane
#define B_PER_BLOCK 4           // batches per block (x3 channels = 12 planes)

// ---------------------------------------------------------------------------
// Kernel A: build per-pixel affine transform  out = T*x + S_c
//   T  = prod_i (1 - a_i)
//   S_c accumulated via s = (1-a)*s + a*ncol  (matches sequential 9-dot blend)
// 1 MB of SoA tables in d_ws; L2-resident, reused 384x by kernel B.
// ---------------------------------------------------------------------------
__global__ void __launch_bounds__(256)
imagedot_mask_kernel(const float* __restrict__ center,  // [9,2]
                     const float* __restrict__ color,   // [9,3]
                     float* __restrict__ T,
                     float* __restrict__ S0,
                     float* __restrict__ S1,
                     float* __restrict__ S2) {
    const int hw = blockIdx.x * 256 + threadIdx.x;   // 0..65535
    const float h = (float)(hw >> 8);
    const float w = (float)(hw & 255);

    const float means[3]    = {0.485f, 0.456f, 0.406f};
    const float inv_stds[3] = {1.0f / 0.229f, 1.0f / 0.224f, 1.0f / 0.225f};
    const float inv_r2 = 1.0f / (RADIUS_C * RADIUS_C);

    float t = 1.0f;
    float s0 = 0.0f, s1 = 0.0f, s2 = 0.0f;

#pragma unroll
    for (int i = 0; i < 9; ++i) {
        const float cy = center[2 * i + 0] * 255.0f;
        const float cx = center[2 * i + 1] * 255.0f;
        const float dy = h - cy;
        const float dx = w - cx;
        const float d  = (dy * dy + dx * dx) * inv_r2;
        // d^20 = ((d^4 * d)^2)^2 ; overflow to +inf -> exp(-inf) = 0 (correct limit)
        const float d2  = d * d;
        const float d4  = d2 * d2;
        const float d5  = d4 * d;
        const float d10 = d5 * d5;
        const float d20 = d10 * d10;
        const float mask = __expf(-d20 + 1e-10f);
        const float a  = ALPHA_C * mask;
        const float om = 1.0f - a;

        const float n0 = (color[3 * i + 0] - means[0]) * inv_stds[0];
        const float n1 = (color[3 * i + 1] - means[1]) * inv_stds[1];
        const float n2 = (color[3 * i + 2] - means[2]) * inv_stds[2];

        t  = t * om;
        s0 = om * s0 + a * n0;
        s1 = om * s1 + a * n1;
        s2 = om * s2 + a * n2;
    }

    T[hw]  = t;
    S0[hw] = s0;
    S1[hw] = s1;
    S2[hw] = s2;
}

// ---------------------------------------------------------------------------
// Kernel B: streaming blend at the roofline minimum of 1 NT B128 load +
// 1 NT B128 store per 16 B of stream.
//   - Block fixes a 256-float4 pixel range; lookup (T + S0/S1/S2) is loaded
//     ONCE into 16 VGPRs and reused across 12 (b,c) planes.
//   - Channel loop fully unrolled -> static S_c selection, no cndmasks.
//   - v_pk_fma_f32 packed math via f4 vector expressions.
//   - global_prefetch_b8 (gfx1250) one line per 8 lanes, one batch ahead;
//     trailing OOB prefetches are speculative and silently dropped.
// ---------------------------------------------------------------------------
__global__ void __launch_bounds__(256)
imagedot_blend_kernel(const f4* __restrict__ x4,
                      const f4* __restrict__ T4,   // NPIX4 f4
                      const f4* __restrict__ S4,   // 3 * NPIX4 f4, channel-major
                      f4* __restrict__ out4) {
    const int tid  = threadIdx.x;
    const int hw4  = blockIdx.x * 256 + tid;         // 0..NPIX4-1
    const int b0   = blockIdx.y * B_PER_BLOCK;       // first batch for this block

    // Per-pixel affine transform cached in registers, reused 12x.
    const f4 t  = T4[hw4];
    const f4 s0 = S4[hw4];
    const f4 s1 = S4[NPIX4 + hw4];
    const f4 s2 = S4[2 * NPIX4 + hw4];

    int base = (b0 * 3) * NPIX4 + hw4;               // f4 index of (b0, c=0, hw4)

#pragma unroll
    for (int j = 0; j < B_PER_BLOCK; ++j) {
        // Prefetch next batch's three planes (128B line per 8 lanes).
        if ((tid & 7) == 0) {
            __builtin_prefetch((const void*)(x4 + base + 3 * NPIX4), 0, 0);
            __builtin_prefetch((const void*)(x4 + base + 4 * NPIX4), 0, 0);
            __builtin_prefetch((const void*)(x4 + base + 5 * NPIX4), 0, 0);
        }

        const f4 x0 = __builtin_nontemporal_load(&x4[base]);
        const f4 x1 = __builtin_nontemporal_load(&x4[base + NPIX4]);
        const f4 x2 = __builtin_nontemporal_load(&x4[base + 2 * NPIX4]);

        const f4 r0 = t * x0 + s0;   // contracts to v_pk_fma_f32 pairs
        const f4 r1 = t * x1 + s1;
        const f4 r2 = t * x2 + s2;

        __builtin_nontemporal_store(r0, &out4[base]);
        __builtin_nontemporal_store(r1, &out4[base + NPIX4]);
        __builtin_nontemporal_store(r2, &out4[base + 2 * NPIX4]);

        base += 3 * NPIX4;
    }
}

extern "C" void kernel_launch(void* const* d_in, const int* in_sizes, int n_in,
                              void* d_out, int out_size, void* d_ws, size_t ws_size,
                              hipStream_t stream) {
    const float* x      = (const float*)d_in[0];   // [B,3,256,256]
    const float* center = (const float*)d_in[1];   // [9,2]
    const float* color  = (const float*)d_in[2];   // [9,3]
    float* out = (float*)d_out;

    // Workspace layout (SoA, 16B-aligned): T | S0 | S1 | S2, each NPIX floats.
    float* T  = (float*)d_ws;
    float* S  = T + NPIX;
    float* S0 = S;
    float* S1 = S + NPIX;
    float* S2 = S + 2 * NPIX;

    // Kernel A: 65536 pixels / 256 threads = 256 blocks.
    imagedot_mask_kernel<<<NPIX / 256, 256, 0, stream>>>(center, color, T, S0, S1, S2);

    // Kernel B: grid = (pixel blocks) x (batch chunks).
    const int batches = out_size / (3 * NPIX);                // 128
    dim3 grid(NPIX4 / 256, batches / B_PER_BLOCK);            // (64, 32)
    imagedot_blend_kernel<<<grid, 256, 0, stream>>>(
        (const f4*)x, (const f4*)T, (const f4*)S, (f4*)out);
}